// Detector_1030792151455
// MI455X (gfx1250) — compile-verified
//
#include <hip/hip_runtime.h>
#include <stdint.h>

// ---------------------------------------------------------------------------
// RetinaNet-style detection loss for MI455X (gfx1250, wave32).
// CDNA5-specific paths used:
//   * global_load_async_to_lds_b128/b32 + s_wait_asynccnt  (async LDS staging)
//   * global_prefetch_b8 via __builtin_prefetch            (pred_cls stream)
// VALU-bound workload (~3G lane-ops vs 126MB HBM) -> IoU uses v_rcp_f32
// instead of IEEE division (self-consistent across both matching passes).
// ---------------------------------------------------------------------------

#define GT_MAX 128   // N = 100 in the reference

// IoU must be bitwise identical between the hq pass and the matching pass so
// that the reference's exact-equality "highest quality match" rule holds.
// => no FMA contraction, fixed operation order, same rcp-based division.
__device__ __forceinline__ float iou_xyxy(const float4 a, const float4 b) {
#pragma clang fp contract(off)
  float lx = fmaxf(a.x, b.x);
  float ly = fmaxf(a.y, b.y);
  float rx = fminf(a.z, b.z);
  float ry = fminf(a.w, b.w);
  float w  = fmaxf(rx - lx, 0.0f);
  float h  = fmaxf(ry - ly, 0.0f);
  float inter = w * h;
  float aa = (a.z - a.x) * (a.w - a.y);
  float ab = (b.z - b.x) * (b.w - b.y);
  float den = aa + ab - inter + 1e-6f;           // >= 1e-6 > 0
  return inter * __builtin_amdgcn_rcpf(den);     // v_rcp_f32 + v_mul_f32
}

__device__ __forceinline__ uint32_t lds_addr_of(const void* p) {
  // Generic (flat) pointers to LDS carry the LDS byte offset in the low 32 bits.
  return (uint32_t)(uintptr_t)p;
}

__device__ __forceinline__ void async_ld_b128(uint32_t lds_addr, const void* gptr) {
  asm volatile("global_load_async_to_lds_b128 %0, %1, off"
               :: "v"(lds_addr), "v"((uint64_t)(uintptr_t)gptr) : "memory");
}
__device__ __forceinline__ void async_ld_b32(uint32_t lds_addr, const void* gptr) {
  asm volatile("global_load_async_to_lds_b32 %0, %1, off"
               :: "v"(lds_addr), "v"((uint64_t)(uintptr_t)gptr) : "memory");
}
__device__ __forceinline__ void async_wait_all() {
  asm volatile("s_wait_asynccnt 0" ::: "memory");
}

__device__ __forceinline__ float focal_term(float x, bool pos) {
  // log_sigmoid(x)  = min(x,0)  - log(1+exp(-|x|))
  // log_sigmoid(-x) = min(-x,0) - log(1+exp(-|x|))
  float ax = fabsf(x);
  float z  = __expf(-ax);
  float opz = 1.0f + z;
  float lg = __logf(opz);
  float r  = __builtin_amdgcn_rcpf(opz);
  float p  = (x >= 0.0f) ? r : z * r;            // sigmoid(x)
  if (pos) {
    float q = 1.0f - p;
    return -0.25f * (q * q) * (fminf(x, 0.0f) - lg);
  }
  return -0.75f * (p * p) * (fminf(-x, 0.0f) - lg);
}

// ---------------------------------------------------------------------------
__global__ void k_init(unsigned* hq_bits, double* loss_sum, unsigned* pos_cnt,
                       int B, int N) {
  int i = blockIdx.x * blockDim.x + threadIdx.x;
  if (i < B * N) hq_bits[i] = 0u;
  if (i < B) { loss_sum[i] = 0.0; pos_cnt[i] = 0u; }
}

// ---------------------------------------------------------------------------
// Pass 1: hq[g] = max over anchors of IoU(gt g, anchor)  (valid gts only)
__global__ void __launch_bounds__(256)
k_hq(const float* __restrict__ acr, const int* __restrict__ lcls,
     const float* __restrict__ lreg, unsigned* __restrict__ hq_bits,
     int AN, int N) {
  __shared__ float4 s_box[GT_MAX];
  __shared__ int    s_cls[GT_MAX];
  __shared__ float  s_wmax[GT_MAX * 8];

  const int b = blockIdx.y;
  const int t = threadIdx.x;
  const int a = blockIdx.x * 256 + t;

  if (t < N) {
    async_ld_b128(lds_addr_of(&s_box[t]), lreg + ((size_t)b * N + t) * 4);
    async_ld_b32 (lds_addr_of(&s_cls[t]), lcls + (size_t)b * N + t);
  }
  async_wait_all();
  __syncthreads();

  float4 abox = make_float4(0.f, 0.f, 0.f, 0.f);   // zero box -> IoU 0 (no-op for max)
  if (a < AN) abox = *(const float4*)(acr + ((size_t)b * AN + a) * 4);

  const int wid  = t >> 5;
  const int lane = t & 31;
  for (int g = 0; g < N; ++g) {
    float v = 0.0f;
    if (s_cls[g] > 0) v = iou_xyxy(s_box[g], abox);
    for (int o = 16; o > 0; o >>= 1) v = fmaxf(v, __shfl_down(v, o, 32));
    if (lane == 0) s_wmax[g * 8 + wid] = v;
  }
  __syncthreads();

  if (t < N && s_cls[t] > 0) {
    float m = s_wmax[t * 8];
    for (int w = 1; w < 8; ++w) m = fmaxf(m, s_wmax[t * 8 + w]);
    atomicMax(&hq_bits[(size_t)b * N + t], __float_as_uint(m));  // IoU >= 0
  }
}

// ---------------------------------------------------------------------------
// Pass 2: matching + focal + regression loss, accumulated per image.
__global__ void __launch_bounds__(256)
k_loss(const float* __restrict__ pcls, const float* __restrict__ preg,
       const float* __restrict__ acr,  const int* __restrict__ lcls,
       const float* __restrict__ lreg, const unsigned* __restrict__ hq_bits,
       double* __restrict__ loss_sum, unsigned* __restrict__ pos_cnt,
       int AN, int N, int C) {
  __shared__ float4 s_box[GT_MAX];
  __shared__ int    s_cls[GT_MAX];
  __shared__ float  s_hq[GT_MAX];
  __shared__ unsigned char s_negpos[256];
  __shared__ unsigned char s_tgt[256];

  const int b     = blockIdx.y;
  const int t     = threadIdx.x;
  const int aBase = blockIdx.x * 256;
  const int a     = aBase + t;

  if (t < N) {
    async_ld_b128(lds_addr_of(&s_box[t]), lreg + ((size_t)b * N + t) * 4);
    async_ld_b32 (lds_addr_of(&s_cls[t]), lcls + (size_t)b * N + t);
    async_ld_b32 (lds_addr_of(&s_hq[t]),  hq_bits + (size_t)b * N + t);
  }
  async_wait_all();
  __syncthreads();

  float best = -1.0f;   // padded gt rows are -1 in the reference
  int   idx  = 0;
  bool  upd = false, mpos = false, mneg = false;
  int   tgt = 0;
  float acc = 0.0f;

  if (a < AN) {
    const float4 abox = *(const float4*)(acr + ((size_t)b * AN + a) * 4);
    for (int g = 0; g < N; ++g) {
      if (s_cls[g] > 0) {
        float v = iou_xyxy(s_box[g], abox);      // bit-identical to pass 1
        upd = upd || (v == s_hq[g]);
        if (v > best) { best = v; idx = g; }     // strict '>' == argmax first-tie
      }
    }
    mneg = (best < 0.4f) && (!upd);
    mpos = (best >= 0.5f) || upd;
    tgt  = mneg ? 0 : s_cls[idx];
    if (mpos) {
      const float4 pbox = *(const float4*)(preg + ((size_t)b * AN + a) * 4);
      float ei = iou_xyxy(pbox, s_box[idx]);
      acc += -__logf(fmaxf(ei, 1e-6f));          // regression IoU loss
    }
  }
  s_negpos[t] = (mneg || mpos) ? 1 : 0;
  s_tgt[t]    = (unsigned char)tgt;

  unsigned long long bal = __ballot(mpos);
  if ((t & 31) == 0) {
    unsigned c = (unsigned)__popcll(bal);
    if (c) atomicAdd(&pos_cnt[b], c);
  }
  __syncthreads();

  // Coalesced focal-loss sweep over this block's [nA x C] logits.
  const int nA = (AN - aBase < 256) ? (AN - aBase) : 256;
  const float* base = pcls + ((size_t)b * AN + (size_t)aBase) * C;

  if ((C & 3) == 0) {
    const int C4 = C >> 2;
    const int total4 = nA * C4;
    const float4* base4 = (const float4*)base;
    for (int j = t; j < total4; j += 256) {
      __builtin_prefetch((const void*)(base4 + j + 2048), 0, 1);  // global_prefetch_b8
      int al = j / C4;
      if (s_negpos[al]) {
        float4 xv = base4[j];
        int c0 = (j - al * C4) << 2;
        int tg = (int)s_tgt[al];
        acc += focal_term(xv.x, tg == c0 + 1);
        acc += focal_term(xv.y, tg == c0 + 2);
        acc += focal_term(xv.z, tg == c0 + 3);
        acc += focal_term(xv.w, tg == c0 + 4);
      }
    }
  } else {
    const int total = nA * C;
    for (int i = t; i < total; i += 256) {
      int al = i / C;
      if (s_negpos[al]) {
        int c = i - al * C;
        acc += focal_term(base[i], (int)s_tgt[al] == c + 1);
      }
    }
  }

  for (int o = 16; o > 0; o >>= 1) acc += __shfl_down(acc, o, 32);
  if ((t & 31) == 0) atomicAdd(&loss_sum[b], (double)acc);
}

// ---------------------------------------------------------------------------
__global__ void k_final(const double* loss_sum, const unsigned* pos_cnt,
                        float* out, int B) {
  int i = threadIdx.x;
  if (i < B) {
    unsigned np = pos_cnt[i];
    if (np < 1u) np = 1u;
    out[i] = (float)(loss_sum[i] / (double)np);
  }
}

// ---------------------------------------------------------------------------
extern "C" void kernel_launch(void* const* d_in, const int* in_sizes, int n_in,
                              void* d_out, int out_size, void* d_ws, size_t ws_size,
                              hipStream_t stream) {
  const float* pcls = (const float*)d_in[1];   // (B, AN, C) logits
  const float* preg = (const float*)d_in[2];   // (B, AN, 4)
  const float* pacr = (const float*)d_in[3];   // (B, AN, 4)
  const int*   lcls = (const int*)d_in[4];     // (B, N)
  const float* lreg = (const float*)d_in[5];   // (B, N, 4)
  float* out = (float*)d_out;

  int B  = in_sizes[0] / 5;                    // locations is (B, 5)
  int N  = in_sizes[4] / B;
  int AN = in_sizes[3] / (4 * B);
  int C  = in_sizes[1] / (B * AN);
  if (N > GT_MAX) N = GT_MAX;

  unsigned char* ws = (unsigned char*)d_ws;
  unsigned* hq_bits = (unsigned*)ws;
  size_t off = ((size_t)B * N * sizeof(unsigned) + 7) & ~(size_t)7;
  double*   loss_sum = (double*)(ws + off);
  off += (size_t)B * sizeof(double);
  unsigned* pos_cnt  = (unsigned*)(ws + off);

  int initN = B * N;
  k_init<<<(initN + 255) / 256, 256, 0, stream>>>(hq_bits, loss_sum, pos_cnt, B, N);

  dim3 grid((AN + 255) / 256, B);
  k_hq  <<<grid, 256, 0, stream>>>(pacr, lcls, lreg, hq_bits, AN, N);
  k_loss<<<grid, 256, 0, stream>>>(pcls, preg, pacr, lcls, lreg, hq_bits,
                                   loss_sum, pos_cnt, AN, N, C);
  k_final<<<1, 256, 0, stream>>>(loss_sum, pos_cnt, out, B);
}